// LocalSpectralProbe_67250597921255
// MI455X (gfx1250) — compile-verified
//
#include <hip/hip_runtime.h>
#include <math.h>
#include <stdint.h>

#define N_    100000
#define DEG_  16
#define E2_   (2 * N_ * DEG_)
#define Q_    16
#define K_    15
#define TOL_  1e-10f
#define EIGTOL_ 1e-6f
#define TB_   256
#define PCHUNK 1024

typedef __attribute__((ext_vector_type(2))) float v2f;
typedef __attribute__((ext_vector_type(8))) float v8f;

// ------------------------- BFS (bitmask per node, one bit per query) ------
__global__ void k_seed(const int* __restrict__ qn, unsigned* __restrict__ active) {
  int q = blockIdx.x * blockDim.x + threadIdx.x;
  if (q < Q_) atomicOr(&active[qn[q]], 1u << q);
}

__global__ void k_hop(const int* __restrict__ rows, const int* __restrict__ cols,
                      const unsigned* __restrict__ active, unsigned* __restrict__ reached) {
  int e = blockIdx.x * blockDim.x + threadIdx.x;
  if (e >= E2_) return;
  unsigned m = active[rows[e]];
  if (m) atomicOr(&reached[cols[e]], m);
}

__global__ void k_merge(unsigned* __restrict__ active, const unsigned* __restrict__ reached) {
  int n = blockIdx.x * blockDim.x + threadIdx.x;
  if (n < N_) active[n] |= reached[n];
}

// ---------------- edge activity mask + weighted degree ---------------------
__global__ void k_edgeprep(const int* __restrict__ rows, const int* __restrict__ cols,
                           const float* __restrict__ vals, const unsigned* __restrict__ active,
                           unsigned* __restrict__ emask, float* __restrict__ deg) {
  int e = blockIdx.x * blockDim.x + threadIdx.x;
  if (e >= E2_) return;
  int r = rows[e], c = cols[e];
  unsigned em = active[r] & active[c];
  emask[e] = em;
  if (em) {
    float wgt = fabsf(vals[e]);
    while (em) {
      int q = __ffs(em) - 1; em &= em - 1;
      atomicAdd(&deg[(size_t)q * N_ + r], wgt);
    }
  }
}

__global__ void k_mcnt(const unsigned* __restrict__ active, float* __restrict__ mcnt) {
  int n = blockIdx.x * blockDim.x + threadIdx.x;
  if (n >= N_) return;
  unsigned m = active[n];
  while (m) { int q = __ffs(m) - 1; m &= m - 1; atomicAdd(&mcnt[q], 1.0f); }
}

// ------------------------- elementwise / reductions ------------------------
__global__ void k_v0init(float* __restrict__ V0, const float* __restrict__ v0,
                         const unsigned* __restrict__ active) {
  int q = blockIdx.y, n = blockIdx.x * blockDim.x + threadIdx.x;
  if (n >= N_) return;
  V0[(size_t)q * N_ + n] = ((active[n] >> q) & 1u) ? v0[n] : 0.0f;
}

__global__ void k_sum(const float* __restrict__ a, float* __restrict__ out) {
  __shared__ float sd[TB_];
  int q = blockIdx.y, n = blockIdx.x * blockDim.x + threadIdx.x;
  float v = (n < N_) ? a[(size_t)q * N_ + n] : 0.0f;
  sd[threadIdx.x] = v; __syncthreads();
  for (int s = TB_ / 2; s > 0; s >>= 1) {
    if (threadIdx.x < s) sd[threadIdx.x] += sd[threadIdx.x + s];
    __syncthreads();
  }
  if (threadIdx.x == 0) atomicAdd(&out[q], sd[0]);
}

__global__ void k_dot(const float* __restrict__ a, const float* __restrict__ b,
                      float* __restrict__ out) {
  __shared__ float sd[TB_];
  int q = blockIdx.y, n = blockIdx.x * blockDim.x + threadIdx.x;
  size_t i = (size_t)q * N_ + n;
  float v = (n < N_) ? a[i] * b[i] : 0.0f;
  sd[threadIdx.x] = v; __syncthreads();
  for (int s = TB_ / 2; s > 0; s >>= 1) {
    if (threadIdx.x < s) sd[threadIdx.x] += sd[threadIdx.x + s];
    __syncthreads();
  }
  if (threadIdx.x == 0) atomicAdd(&out[q], sd[0]);
}

__global__ void k_center(float* __restrict__ w, const unsigned* __restrict__ active,
                         const float* __restrict__ s, const float* __restrict__ mcnt) {
  int q = blockIdx.y, n = blockIdx.x * blockDim.x + threadIdx.x;
  if (n >= N_) return;
  size_t i = (size_t)q * N_ + n;
  w[i] = ((active[n] >> q) & 1u) ? (w[i] - s[q] / fmaxf(mcnt[q], 1.0f)) : 0.0f;
}

__global__ void k_normalize(float* __restrict__ v, const float* __restrict__ s2) {
  int q = blockIdx.y, n = blockIdx.x * blockDim.x + threadIdx.x;
  if (n >= N_) return;
  v[(size_t)q * N_ + n] /= fmaxf(sqrtf(s2[q]), TOL_);
}

__global__ void k_mv_init(float* __restrict__ w, const float* __restrict__ deg,
                          const float* __restrict__ x) {
  int q = blockIdx.y, n = blockIdx.x * blockDim.x + threadIdx.x;
  if (n >= N_) return;
  size_t i = (size_t)q * N_ + n;
  w[i] = deg[i] * x[i];
}

// Laplacian SpMV: w[q][r] -= |val| * x[q][c] for every active (edge,query)
__global__ void k_spmv(const int* __restrict__ rows, const int* __restrict__ cols,
                       const float* __restrict__ vals, const unsigned* __restrict__ emask,
                       const float* __restrict__ x, float* __restrict__ w) {
  int e = blockIdx.x * blockDim.x + threadIdx.x;
  if (e >= E2_) return;
  unsigned em = emask[e];
  if (!em) return;
  float wgt = fabsf(vals[e]);
  int r = rows[e], c = cols[e];
  while (em) {
    int q = __ffs(em) - 1; em &= em - 1;
    atomicAdd(&w[(size_t)q * N_ + r], -wgt * x[(size_t)q * N_ + c]);
  }
}

__global__ void k_sub2(float* __restrict__ w, const float* __restrict__ Vi,
                       const float* __restrict__ a, const float* __restrict__ Vim1,
                       const float* __restrict__ b) {
  int q = blockIdx.y, n = blockIdx.x * blockDim.x + threadIdx.x;
  if (n >= N_) return;
  size_t i = (size_t)q * N_ + n;
  float val = w[i] - a[q] * Vi[i];
  if (b) val -= b[q] * Vim1[i];
  w[i] = val;
}

// ---- full reorthogonalization coefficients via V_WMMA_F32_16X16X4_F32 ----
// One wave per (n-slice, q). A(16x4): rows = j (Lanczos vector index, zero-
// padded past nj), cols = 4 consecutive n. B(4x16): w[n] broadcast across all
// 16 columns. Chained f32 accumulator; column 0 of D = partial coef[0..15].
__global__ void k_proj_coef(const float* __restrict__ V, const float* __restrict__ w,
                            float* __restrict__ coef, int nj) {
  int q = blockIdx.y;
  int lane = threadIdx.x;            // wave32: 0..31
  int j = lane & 15;                 // A row (M)
  int kb = (lane >> 4) << 1;         // lanes 0-15 hold K=0,1; lanes 16-31 K=2,3
  float jscale = (j < nj) ? 1.0f : 0.0f;   // zero-pad invalid rows, keep EXEC full
  const float* vj = V + ((size_t)(j < nj ? j : 0) * Q_ + q) * N_;
  const float* wq = w + (size_t)q * N_;
  int n0 = blockIdx.x * PCHUNK;
  int n1 = n0 + PCHUNK; if (n1 > N_) n1 = N_;   // N_ and PCHUNK are multiples of 4
  v8f acc = {};
  #pragma unroll 4
  for (int n = n0; n < n1; n += 4) {
    v2f a, b;
    a.x = vj[n + kb] * jscale;
    a.y = vj[n + kb + 1] * jscale;
    b.x = wq[n + kb];
    b.y = wq[n + kb + 1];
    acc = __builtin_amdgcn_wmma_f32_16x16x4_f32(false, a, false, b, (short)0, acc,
                                                false, false);
  }
  // D column 0 lives in lane 0 (M=0..7 in VGPR 0..7) and lane 16 (M=8..15)
  if (lane == 0) {
    for (int r = 0; r < 8; ++r) if (r < nj) atomicAdd(&coef[r * Q_ + q], acc[r]);
  } else if (lane == 16) {
    for (int r = 0; r < 8; ++r) if (8 + r < nj) atomicAdd(&coef[(8 + r) * Q_ + q], acc[r]);
  }
}

__global__ void k_proj_update(float* __restrict__ w, const float* __restrict__ V,
                              const float* __restrict__ coef, int nj) {
  int q = blockIdx.y, n = blockIdx.x * blockDim.x + threadIdx.x;
  if (n >= N_) return;
  size_t i = (size_t)q * N_ + n;
  float val = w[i];
  for (int j = 0; j < nj; ++j)
    val -= coef[j * Q_ + q] * V[((size_t)j * Q_ + q) * N_ + n];
  w[i] = val;
}

__global__ void k_beta(const float* __restrict__ s2, float* __restrict__ beta_i) {
  int q = blockIdx.x * blockDim.x + threadIdx.x;
  if (q < Q_) beta_i[q] = sqrtf(s2[q]);
}

__global__ void k_scalefrom(float* __restrict__ dst, const float* __restrict__ src,
                            const float* __restrict__ s2) {
  int q = blockIdx.y, n = blockIdx.x * blockDim.x + threadIdx.x;
  if (n >= N_) return;
  size_t i = (size_t)q * N_ + n;
  dst[i] = src[i] / fmaxf(sqrtf(s2[q]), TOL_);
}

// ------------- per-query 15x15 tridiagonal eigensolve (TQL2) ---------------
__global__ void k_eig(const float* __restrict__ alphas, const float* __restrict__ betas,
                      float* __restrict__ y, float* __restrict__ lam2) {
  int q = threadIdx.x;
  if (q >= Q_) return;
  float d[K_], e[K_], z[K_ * K_];
  for (int i = 0; i < K_; ++i) {
    d[i] = alphas[i * Q_ + q];
    e[i] = (i < K_ - 1) ? betas[i * Q_ + q] : 0.0f;
  }
  for (int i = 0; i < K_ * K_; ++i) z[i] = 0.0f;
  for (int i = 0; i < K_; ++i) z[i * K_ + i] = 1.0f;

  for (int l = 0; l < K_; ++l) {
    int iter = 0, m;
    do {
      for (m = l; m < K_ - 1; ++m) {
        float dd = fabsf(d[m]) + fabsf(d[m + 1]);
        if (fabsf(e[m]) <= 1.19209290e-07f * dd) break;
      }
      if (m != l) {
        if (iter++ >= 50) break;
        float g = (d[l + 1] - d[l]) / (2.0f * e[l]);
        float r = sqrtf(g * g + 1.0f);
        g = d[m] - d[l] + e[l] / (g + copysignf(r, g));
        float s = 1.0f, c = 1.0f, p = 0.0f;
        for (int i = m - 1; i >= l; --i) {
          float f = s * e[i], b = c * e[i];
          r = sqrtf(f * f + g * g);
          e[i + 1] = r;
          if (r == 0.0f) { d[i + 1] -= p; e[m] = 0.0f; break; }
          s = f / r; c = g / r;
          g = d[i + 1] - p;
          r = (d[i] - g) * s + 2.0f * c * b;
          p = s * r; d[i + 1] = g + p; g = c * r - b;
          for (int k = 0; k < K_; ++k) {
            f = z[k * K_ + i + 1];
            z[k * K_ + i + 1] = s * z[k * K_ + i] + c * f;
            z[k * K_ + i]     = c * z[k * K_ + i] - s * f;
          }
        }
        d[l] -= p; e[l] = g; e[m] = 0.0f;
      }
    } while (m != l);
  }
  // smallest eigenvalue above EIG_TOL (== first valid index of sorted eigh)
  int best = 0; float bl = 0.0f; bool found = false;
  for (int i = 0; i < K_; ++i)
    if (d[i] > EIGTOL_ && (!found || d[i] < bl)) { best = i; bl = d[i]; found = true; }
  if (!found) bl = d[0];
  lam2[q] = bl;
  for (int k = 0; k < K_; ++k) y[q * K_ + k] = z[k * K_ + best];
}

__global__ void k_fiedler(float* __restrict__ out, const float* __restrict__ V,
                          const float* __restrict__ y) {
  int q = blockIdx.y, n = blockIdx.x * blockDim.x + threadIdx.x;
  if (n >= N_) return;
  float acc = 0.0f;
  #pragma unroll
  for (int k = 0; k < K_; ++k)
    acc += V[((size_t)k * Q_ + q) * N_ + n] * y[q * K_ + k];
  out[(size_t)q * N_ + n] = acc;
}

// --------------------------------- host ------------------------------------
extern "C" void kernel_launch(void* const* d_in, const int* in_sizes, int n_in,
                              void* d_out, int out_size, void* d_ws, size_t ws_size,
                              hipStream_t stream) {
  const int*   rows = (const int*)d_in[0];
  const int*   cols = (const int*)d_in[1];
  const float* vals = (const float*)d_in[2];
  const int*   qn   = (const int*)d_in[3];
  const float* v0   = (const float*)d_in[4];
  float* out  = (float*)d_out;
  float* lam2 = out + (size_t)Q_ * N_;

  uint8_t* p = (uint8_t*)d_ws;
  auto alloc = [&](size_t bytes) -> void* {
    void* r = (void*)p; p += (bytes + 255) & ~(size_t)255; return r;
  };
  unsigned* active  = (unsigned*)alloc((size_t)N_ * 4);
  unsigned* reached = (unsigned*)alloc((size_t)N_ * 4);
  unsigned* emask   = (unsigned*)alloc((size_t)E2_ * 4);
  float* deg    = (float*)alloc((size_t)Q_ * N_ * 4);
  float* V      = (float*)alloc((size_t)(K_ + 1) * Q_ * N_ * 4);
  float* w      = (float*)alloc((size_t)Q_ * N_ * 4);
  float* mcnt   = (float*)alloc(Q_ * 4);
  float* alphas = (float*)alloc(K_ * Q_ * 4);
  float* betas  = (float*)alloc(K_ * Q_ * 4);
  float* coef   = (float*)alloc(16 * Q_ * 4);
  float* s1     = (float*)alloc(Q_ * 4);
  float* yv     = (float*)alloc(Q_ * K_ * 4);

  const int EB = (E2_ + TB_ - 1) / TB_;
  const int NB = (N_ + TB_ - 1) / TB_;
  dim3 gQN(NB, Q_);
  dim3 gP((N_ + PCHUNK - 1) / PCHUNK, Q_);

  hipMemsetAsync(active, 0, (size_t)N_ * 4, stream);
  hipMemsetAsync(deg,    0, (size_t)Q_ * N_ * 4, stream);
  hipMemsetAsync(mcnt,   0, Q_ * 4, stream);
  hipMemsetAsync(alphas, 0, K_ * Q_ * 4, stream);
  hipMemsetAsync(betas,  0, K_ * Q_ * 4, stream);

  // 2-hop BFS
  k_seed<<<1, 32, 0, stream>>>(qn, active);
  for (int h = 0; h < 2; ++h) {
    hipMemsetAsync(reached, 0, (size_t)N_ * 4, stream);
    k_hop<<<EB, TB_, 0, stream>>>(rows, cols, active, reached);
    k_merge<<<NB, TB_, 0, stream>>>(active, reached);
  }

  // local Laplacian pieces
  k_edgeprep<<<EB, TB_, 0, stream>>>(rows, cols, vals, active, emask, deg);
  k_mcnt<<<NB, TB_, 0, stream>>>(active, mcnt);

  // v = (v0*af - mean)*af, normalized
  k_v0init<<<gQN, TB_, 0, stream>>>(V, v0, active);
  hipMemsetAsync(s1, 0, Q_ * 4, stream);
  k_sum<<<gQN, TB_, 0, stream>>>(V, s1);
  k_center<<<gQN, TB_, 0, stream>>>(V, active, s1, mcnt);
  hipMemsetAsync(s1, 0, Q_ * 4, stream);
  k_dot<<<gQN, TB_, 0, stream>>>(V, V, s1);
  k_normalize<<<gQN, TB_, 0, stream>>>(V, s1);

  // Lanczos with full reorthogonalization
  for (int i = 0; i < K_; ++i) {
    const float* Vi = V + (size_t)i * Q_ * N_;
    float*       Vn = V + (size_t)(i + 1) * Q_ * N_;
    k_mv_init<<<gQN, TB_, 0, stream>>>(w, deg, Vi);
    k_spmv<<<EB, TB_, 0, stream>>>(rows, cols, vals, emask, Vi, w);
    k_dot<<<gQN, TB_, 0, stream>>>(Vi, w, alphas + i * Q_);
    k_sub2<<<gQN, TB_, 0, stream>>>(w, Vi, alphas + i * Q_,
        (i > 0) ? V + (size_t)(i - 1) * Q_ * N_ : Vi,
        (i > 0) ? betas + (i - 1) * Q_ : (const float*)nullptr);
    hipMemsetAsync(coef, 0, 16 * Q_ * 4, stream);
    k_proj_coef<<<gP, 32, 0, stream>>>(V, w, coef, i + 1);      // WMMA
    k_proj_update<<<gQN, TB_, 0, stream>>>(w, V, coef, i + 1);
    hipMemsetAsync(s1, 0, Q_ * 4, stream);
    k_sum<<<gQN, TB_, 0, stream>>>(w, s1);
    k_center<<<gQN, TB_, 0, stream>>>(w, active, s1, mcnt);
    hipMemsetAsync(s1, 0, Q_ * 4, stream);
    k_dot<<<gQN, TB_, 0, stream>>>(w, w, s1);
    k_beta<<<1, 32, 0, stream>>>(s1, betas + i * Q_);
    k_scalefrom<<<gQN, TB_, 0, stream>>>(Vn, w, s1);
  }

  k_eig<<<1, 32, 0, stream>>>(alphas, betas, yv, lam2);
  k_fiedler<<<gQN, TB_, 0, stream>>>(out, V, yv);
}